// KVCache_57690000720265
// MI455X (gfx1250) — compile-verified
//
#include <hip/hip_runtime.h>
#include <hip/hip_bf16.h>
#include <stdint.h>

// Problem geometry (fixed by the reference's setup_inputs)
#define LAYERS  16
#define BSZ     4
#define MAX_SEQ 4096
#define KVH     8
#define HD      128
#define INSERT  16

typedef uint32_t u32;
typedef unsigned long long u64;
typedef unsigned int uv4 __attribute__((ext_vector_type(4)));   // 16B = 8 bf16

#define CACHE_ELEMS ((u64)LAYERS * BSZ * MAX_SEQ * KVH * HD)    // 268,435,456 bf16 per cache

// ---- fp32 -> bf16 round-to-nearest-even, pack 8 values into 16 bytes ----
__device__ __forceinline__ uv4 cvt8_f32_bf16(const float* f) {
    u32 h[8];
#pragma unroll
    for (int i = 0; i < 8; ++i) {
        union { float f; u32 u; } c; c.f = f[i];
        u32 x = c.u;
        x += 0x7fffu + ((x >> 16) & 1u);   // RNE
        h[i] = x >> 16;
    }
    uv4 o;
    o.x = h[0] | (h[1] << 16);
    o.y = h[2] | (h[3] << 16);
    o.z = h[4] | (h[5] << 16);
    o.w = h[6] | (h[7] << 16);
    return o;
}

// =====================================================================
// Kernel 1: bulk cache copy k_cache->k_new, v_cache->v_new through the
// CDNA5 async DMA path: global_load_async_to_lds_b128 (8-deep per lane)
// -> s_wait_asynccnt -> global_store_async_from_lds_b128. 32KB LDS/block.
// =====================================================================
#define CPY_DEPTH 8
__global__ void bulk_copy_async(const __hip_bfloat16* __restrict__ kc,
                                const __hip_bfloat16* __restrict__ vc,
                                __hip_bfloat16* __restrict__ out,
                                const int* __restrict__ nrep_p) {
    __shared__ alignas(16) unsigned char smem[256 * CPY_DEPTH * 16];
    const int   nrep = *nrep_p;
    const u64   K1   = (u64)BSZ * MAX_SEQ * KVH * (u64)nrep * HD;   // keys elem count
    const u64   NVEC = CACHE_ELEMS / 8;                              // 16B vectors per cache
    const unsigned char* src = (const unsigned char*)(blockIdx.y ? vc : kc);
    unsigned char* dst = (unsigned char*)(out + 2 * K1 + (u64)blockIdx.y * CACHE_ELEMS);

    // raw LDS byte offset of smem (single static LDS allocation)
    unsigned lds0 = (unsigned)(u64)(__attribute__((address_space(3))) unsigned char*)smem;
    const unsigned t = threadIdx.x;

    const u64 chunk  = (u64)256 * CPY_DEPTH;             // vectors per block-iter
    const u64 stride = (u64)gridDim.x * chunk;
    for (u64 base = (u64)blockIdx.x * chunk; base < NVEC; base += stride) {
#pragma unroll
        for (int s = 0; s < CPY_DEPTH; ++s) {
            u64 idx = base + (u64)s * 256 + t;
            if (idx < NVEC) {
                unsigned laddr = lds0 + (unsigned)(s * 256 + t) * 16u;
                const unsigned char* g = src + idx * 16;
                asm volatile("global_load_async_to_lds_b128 %0, %1, off"
                             :: "v"(laddr), "v"(g) : "memory");
            }
        }
        asm volatile("s_wait_asynccnt 0" ::: "memory");
#pragma unroll
        for (int s = 0; s < CPY_DEPTH; ++s) {
            u64 idx = base + (u64)s * 256 + t;
            if (idx < NVEC) {
                unsigned laddr = lds0 + (unsigned)(s * 256 + t) * 16u;
                unsigned char* g = dst + idx * 16;
                asm volatile("global_store_async_from_lds_b128 %0, %1, off"
                             :: "v"(g), "v"(laddr) : "memory");
            }
        }
        asm volatile("s_wait_asynccnt 0" ::: "memory");   // LDS slots reused next iter
    }
}

// =====================================================================
// Kernel 2: convert fp32 xk/xv -> bf16 and scatter into the just-copied
// cache at [layer, b, cur_pos+i, h, :]. Tiny (128KB of writes).
// =====================================================================
__global__ void insert_kernel(const float* __restrict__ xk,
                              const float* __restrict__ xv,
                              __hip_bfloat16* __restrict__ out,
                              const int* li_p, const int* cp_p, const int* nr_p) {
    const int layer = *li_p, cur = *cp_p, nrep = *nr_p;
    const u64 K1   = (u64)BSZ * MAX_SEQ * KVH * (u64)nrep * HD;
    __hip_bfloat16* dst = out + 2 * K1 + (u64)blockIdx.y * CACHE_ELEMS;
    const float* src = blockIdx.y ? xv : xk;

    const int NIT = BSZ * INSERT * KVH * (HD / 8);   // 8192 16B items
    for (int v = blockIdx.x * blockDim.x + threadIdx.x; v < NIT;
         v += gridDim.x * blockDim.x) {
        int d = (v % (HD / 8)) * 8;
        int h = (v / (HD / 8)) % KVH;
        int i = (v / ((HD / 8) * KVH)) % INSERT;
        int b =  v / ((HD / 8) * KVH * INSERT);
        float f[8];
        const float* sp = src + (((u64)(b * INSERT + i) * KVH + h) * HD + d);
#pragma unroll
        for (int j = 0; j < 8; ++j) f[j] = sp[j];
        uv4 pk = cvt8_f32_bf16(f);
        u64 o = ((((u64)layer * BSZ + b) * MAX_SEQ + (u64)(cur + i)) * KVH + h) * HD + d;
        *(uv4*)(dst + o) = pk;
    }
}

// =====================================================================
// Kernel 3: GQA expansion. keys[b,s,h*nrep+r,:] = updated cache row
// (insert rows sourced directly from fp32 inputs, rest from bf16 cache).
// Streaming NT stores keep 268MB of write-once data out of L2.
// =====================================================================
__global__ void expand_kernel(const float* __restrict__ xk,
                              const float* __restrict__ xv,
                              const __hip_bfloat16* __restrict__ kc,
                              const __hip_bfloat16* __restrict__ vc,
                              __hip_bfloat16* __restrict__ out,
                              const int* li_p, const int* cp_p, const int* nr_p) {
    const int layer = *li_p, cur = *cp_p, nrep = *nr_p;
    const int QH = KVH * nrep;
    const u64 K1 = (u64)BSZ * MAX_SEQ * (u64)QH * HD;
    const float* fsrc = blockIdx.y ? xv : xk;
    const __hip_bfloat16* csrc = blockIdx.y ? vc : kc;
    __hip_bfloat16* dst = out + (u64)blockIdx.y * K1;

    const u64 NIT = (u64)BSZ * MAX_SEQ * KVH * (HD / 8);
    for (u64 v = (u64)blockIdx.x * blockDim.x + threadIdx.x; v < NIT;
         v += (u64)gridDim.x * blockDim.x) {
        int d = (int)(v % (HD / 8)) * 8;
        int h = (int)((v / (HD / 8)) % KVH);
        int s = (int)((v / ((u64)(HD / 8) * KVH)) % MAX_SEQ);
        int b = (int)(v / ((u64)(HD / 8) * KVH * MAX_SEQ));
        uv4 pk;
        if (s >= cur && s < cur + INSERT) {
            int i = s - cur;
            float f[8];
            const float* sp = fsrc + (((u64)(b * INSERT + i) * KVH + h) * HD + d);
#pragma unroll
            for (int j = 0; j < 8; ++j) f[j] = sp[j];
            pk = cvt8_f32_bf16(f);
        } else {
            pk = __builtin_nontemporal_load(
                (const uv4*)(csrc + ((((u64)layer * BSZ + b) * MAX_SEQ + s) * KVH + h) * HD + d));
        }
        __hip_bfloat16* dp = dst + (((u64)b * MAX_SEQ + s) * QH + (u64)h * nrep) * HD + d;
        for (int r = 0; r < nrep; ++r)
            __builtin_nontemporal_store(pk, (uv4*)(dp + (u64)r * HD));
    }
}

// =====================================================================
// Launcher. Inputs (setup_inputs order): xk f32, xv f32, k_cache bf16,
// v_cache bf16, layer_idx i32[1], cur_pos i32[1], n_rep i32[1].
// Output: keys | values | k_new | v_new (bf16, flat-concatenated).
// =====================================================================
extern "C" void kernel_launch(void* const* d_in, const int* in_sizes, int n_in,
                              void* d_out, int out_size, void* d_ws, size_t ws_size,
                              hipStream_t stream) {
    (void)in_sizes; (void)n_in; (void)out_size; (void)d_ws; (void)ws_size;
    const float*          xk = (const float*)d_in[0];
    const float*          xv = (const float*)d_in[1];
    const __hip_bfloat16* kc = (const __hip_bfloat16*)d_in[2];
    const __hip_bfloat16* vc = (const __hip_bfloat16*)d_in[3];
    const int*            li = (const int*)d_in[4];
    const int*            cp = (const int*)d_in[5];
    const int*            nr = (const int*)d_in[6];
    __hip_bfloat16*      out = (__hip_bfloat16*)d_out;

    // 1) full cache copy (HBM->LDS->HBM async DMA), grid.y = {K,V}
    bulk_copy_async<<<dim3(2048, 2), 256, 0, stream>>>(kc, vc, out, nr);
    // 2) overwrite the freshly inserted 16 rows (stream-ordered after the copy)
    insert_kernel<<<dim3(32, 2), 256, 0, stream>>>(xk, xv, out, li, cp, nr);
    // 3) GQA-expanded keys/values (independent of 1 & 2: reads only d_in)
    expand_kernel<<<dim3(4096, 2), 256, 0, stream>>>(xk, xv, kc, vc, out, li, cp, nr);
}